// incremental_HMM_40329742909509
// MI455X (gfx1250) — compile-verified
//
#include <hip/hip_runtime.h>
#include <math.h>

#define T_LEN 2000
#define D_LEN 512
#define R_LEN 64
#define T_TILE 40               // 50 blocks in y cover T=2000
#define LOG_2PI 1.8378770664093453f
#define LOG2E   1.4426950408889634f
#define LN2     0.6931471805599453f

typedef __attribute__((ext_vector_type(2))) float v2f;
typedef __attribute__((ext_vector_type(8))) float v8f;

// Fused DPP-arithmetic butterfly steps: DPP is a src0 modifier on VOP2, so
// max/add with a row_ror:N rotated copy is ONE VALU instruction (vs
// mov_dpp + canonicalize + max when going through the generic builtin).
// row_ror rotates within 16-lane rows => wave32 halves reduce independently.
#define FMAX_ROR16(out, in, N)                                              \
  asm("v_max_num_f32_dpp %0, %1, %1 row_ror:" #N                            \
      " row_mask:0xf bank_mask:0xf bound_ctrl:1"                            \
      : "=&v"(out) : "v"(in))

#define FADD_ROR16(out, in, N)                                              \
  asm("v_add_f32_dpp %0, %1, %1 row_ror:" #N                                \
      " row_mask:0xf bank_mask:0xf bound_ctrl:1"                            \
      : "=&v"(out) : "v"(in))

__device__ __forceinline__ float row_max16(float v) {
  float a, b, c, d;
  FMAX_ROR16(a, v, 1);
  FMAX_ROR16(b, a, 2);
  FMAX_ROR16(c, b, 4);
  FMAX_ROR16(d, c, 8);
  return d;  // every lane of the 16-lane row holds the row max
}

__device__ __forceinline__ float row_sum16(float v) {
  float a, b, c, d;
  FADD_ROR16(a, v, 1);
  FADD_ROR16(b, a, 2);
  FADD_ROR16(c, b, 4);
  FADD_ROR16(d, c, 8);
  return d;  // every lane of the 16-lane row holds the row sum
}

// ---------------------------------------------------------------------------
// Kernel 0: zero the output accumulator (harness poisons it; we atomicAdd).
// ---------------------------------------------------------------------------
__global__ void hmm_zero_out(float* __restrict__ out, int n) {
  int i = blockIdx.x * blockDim.x + threadIdx.x;
  if (i < n) out[i] = 0.0f;
}

// ---------------------------------------------------------------------------
// Kernel 1: serial scan  h_{t+1} = h_t @ P  (one block, 64 threads = 2 waves)
// Emits interleaved per-step coefficients for the rank-3 logit form,
// PRE-SCALED by log2(e) so the hot loop runs in the log2 domain and can use
// raw v_exp_f32 / v_log_f32 (which are natively base-2) with no per-term
// rescaling:
//   logit2[t,d,r] = (A[t,r] + x*B[t,r] + x^2*C[r])            (log2 units)
//   AB[t,r] = log2e * (log_h - 0.5*mu^2/s^2 - log s - 0.5*log2pi,  mu/s^2)
//   C[r]    = log2e * (-0.5/s^2)
// ---------------------------------------------------------------------------
__global__ void hmm_scan_kernel(const float* __restrict__ transition,
                                const float* __restrict__ sig,
                                const float* __restrict__ init_w,
                                const float* __restrict__ mu_rates,
                                float2* __restrict__ ABws,
                                float* __restrict__ Cws) {
  __shared__ float P[R_LEN * R_LEN];
  __shared__ float hs[R_LEN];
  const int r = threadIdx.x;  // 0..63
  for (int i = r; i < R_LEN * R_LEN; i += R_LEN) P[i] = transition[i];
  hs[r] = init_w[r];
  const float s      = sig[r];
  const float inv_s2 = 1.0f / (s * s);
  const float logs   = logf(s);
  const float mu0    = mu_rates[r];
  Cws[r] = -0.5f * inv_s2 * LOG2E;
  __syncthreads();

  for (int t = 0; t < T_LEN; ++t) {
    const float h_r = hs[r];
    float acc = 0.0f, ssum = 0.0f;
    #pragma unroll 8
    for (int k = 0; k < R_LEN; ++k) {
      const float hk = hs[k];
      ssum += hk;
      acc = fmaf(hk, P[k * R_LEN + r], acc);
    }
    const float log_h = logf(h_r) - logf(ssum);
    // mu drift: +10 at every i%100==0 including i=0 -> 10*(t/100 + 1)
    const float mu = mu0 + 10.0f * (float)(t / 100 + 1);
    float2 ab;
    ab.x = (log_h - 0.5f * mu * mu * inv_s2 - logs - 0.5f * LOG_2PI) * LOG2E;
    ab.y = mu * inv_s2 * LOG2E;
    ABws[t * R_LEN + r] = ab;
    __syncthreads();
    hs[r] = acc;
    __syncthreads();
  }
}

// ---------------------------------------------------------------------------
// Kernel 2: one wave per (16-d chunk, 40-t slice).
// logits tile (16 d x 16 r) = WMMA_F32_16X16X4_F32:
//   Amat[d,k] = [1, x, x^2, 0],  Bmat[k,r] = [A_t[r], B_t[r], C[r], 0]
// 4 r-tiles cover R=64; logsumexp (in log2 domain) via register max over
// tiles + fused DPP row_ror butterflies (C/D layout: N=r across lanes within
// each 16-lane half, M=d across the 8 accumulator VGPRs). Loads for t+1 are
// issued before the compute for t so vmem latency hides under the exp chain.
// ---------------------------------------------------------------------------
__global__ void
__launch_bounds__(32)
hmm_lse_kernel(const float* __restrict__ X,
               const float2* __restrict__ ABws,
               const float* __restrict__ Cws,
               float* __restrict__ out) {
  const int lane  = threadIdx.x;      // 0..31 (wave32)
  const int n     = lane & 15;        // N index within half
  const bool lo   = lane < 16;        // lower half: K=0/1, upper: K=2/3
  const int dbase = blockIdx.x * 16;
  const int tbeg  = blockIdx.y * T_TILE;

  float cval[4];
  #pragma unroll
  for (int j = 0; j < 4; ++j) cval[j] = Cws[j * 16 + n];

  float acc[8];
  #pragma unroll
  for (int i = 0; i < 8; ++i) acc[i] = 0.0f;

  // Prologue: loads for t = tbeg.
  float  xc = X[tbeg * D_LEN + dbase + n];
  float2 abc[4];
  #pragma unroll
  for (int j = 0; j < 4; ++j) abc[j] = ABws[tbeg * R_LEN + j * 16 + n];

  for (int t = tbeg; t < tbeg + T_TILE; ++t) {
    // Issue next iteration's loads first (last iteration reloads t: harmless).
    const int tn = (t + 1 < T_LEN) ? t + 1 : t;
    const float  xn = X[tn * D_LEN + dbase + n];
    float2 abn[4];
    #pragma unroll
    for (int j = 0; j < 4; ++j) abn[j] = ABws[tn * R_LEN + j * 16 + n];

    // A-matrix 16x4 f32 layout: lanes 0-15 hold K=0 (v0) / K=1 (v1),
    // lanes 16-31 hold K=2 (v0) / K=3 (v1).
    v2f amat;
    amat.x = lo ? 1.0f : xc * xc;   // K0 = 1,  K2 = x^2
    amat.y = lo ? xc   : 0.0f;      // K1 = x,  K3 = 0

    v8f dt[4];
    #pragma unroll
    for (int j = 0; j < 4; ++j) {
      v2f bmat;
      bmat.x = lo ? abc[j].x : cval[j];  // K0 row = A_t,  K2 row = C
      bmat.y = lo ? abc[j].y : 0.0f;     // K1 row = B_t,  K3 row = 0
      v8f cz = {};
      dt[j] = __builtin_amdgcn_wmma_f32_16x16x4_f32(
          /*neg_a=*/false, amat, /*neg_b=*/false, bmat,
          /*c_mod=*/(short)0, cz, /*reuse_a=*/false, /*reuse_b=*/false);
    }

    // log2-sum-exp2 over 64 r-values per d-row, accumulate over t.
    // v_exp_f32 / v_log_f32 are natively base-2: no per-term scaling needed.
    #pragma unroll
    for (int i = 0; i < 8; ++i) {
      float mx = fmaxf(fmaxf(dt[0][i], dt[1][i]), fmaxf(dt[2][i], dt[3][i]));
      mx = row_max16(mx);
      float se = __builtin_amdgcn_exp2f(dt[0][i] - mx) +
                 __builtin_amdgcn_exp2f(dt[1][i] - mx) +
                 __builtin_amdgcn_exp2f(dt[2][i] - mx) +
                 __builtin_amdgcn_exp2f(dt[3][i] - mx);
      se = row_sum16(se);
      acc[i] += mx + __builtin_amdgcn_logf(se);  // still log2 units
    }

    xc = xn;
    #pragma unroll
    for (int j = 0; j < 4; ++j) abc[j] = abn[j];
  }

  // Lane n==0 of each half holds d = dbase + i (lo) / dbase + 8 + i (hi).
  // Single multiply by ln2 converts the log2-domain sum back to nats.
  if (n == 0) {
    const int doff = dbase + (lo ? 0 : 8);
    #pragma unroll
    for (int i = 0; i < 8; ++i) atomicAdd(&out[doff + i], acc[i] * LN2);
  }
}

// ---------------------------------------------------------------------------
extern "C" void kernel_launch(void* const* d_in, const int* in_sizes, int n_in,
                              void* d_out, int out_size, void* d_ws, size_t ws_size,
                              hipStream_t stream) {
  const float* X          = (const float*)d_in[0];  // [T, D]
  const float* transition = (const float*)d_in[1];  // [R, R]
  const float* sig        = (const float*)d_in[2];  // [1, R]
  const float* init_w     = (const float*)d_in[3];  // [1, R]
  const float* mu_rates   = (const float*)d_in[4];  // [1, R]
  float* out = (float*)d_out;                       // [D]

  const size_t tbl = (size_t)T_LEN * R_LEN * 2 * sizeof(float);  // 1 MB
  float2* ABws = (float2*)d_ws;
  float*  Cws  = (float*)((char*)d_ws + tbl);

  hmm_zero_out<<<(D_LEN + 255) / 256, 256, 0, stream>>>(out, D_LEN);

  hmm_scan_kernel<<<1, R_LEN, 0, stream>>>(transition, sig, init_w, mu_rates,
                                           ABws, Cws);

  dim3 grid(D_LEN / 16, T_LEN / T_TILE);  // 32 x 50 waves
  hmm_lse_kernel<<<grid, 32, 0, stream>>>(X, ABws, Cws, out);
}